// ScaledDotProductAttention_60713657697182
// MI455X (gfx1250) — compile-verified
//
#include <hip/hip_runtime.h>

typedef float v2f __attribute__((ext_vector_type(2)));
typedef float v8f __attribute__((ext_vector_type(8)));

#define NB 4
#define NH 16
#define NS 2048
#define NDK 64

__global__ __launch_bounds__(32)
void sdpa_fp32_wmma_kernel(const float* __restrict__ Q,
                           const float* __restrict__ K,
                           const float* __restrict__ V,
                           const int*   __restrict__ mask,
                           float* __restrict__ out,
                           float* __restrict__ attn)
{
    __shared__ __align__(16) float pbuf[16 * 16];   // one P tile per (single-wave) workgroup

    const int lane = threadIdx.x & 31;
    const int n    = lane & 15;   // column (B,C,D) / row (A) index within half
    const int lh   = lane >> 4;   // lane half

    const int wid = blockIdx.x;           // 0..8191  -> (b, h, q-strip)
    const int qt  = wid & 127;            // q-strip index (16 rows each)
    const int h   = (wid >> 7) & 15;
    const int b   = wid >> 11;

    const size_t bh = (size_t)(b * NH + h);
    const float* Qb = Q + bh * NS * NDK + (size_t)qt * 16 * NDK;
    const float* Kb = K + bh * NS * NDK;
    const float* Vb = V + bh * NS * NDK;
    const int*   Mb = mask + (size_t)b * NS * NS;                 // [B,1,S,S]
    float*       Ob = out  + bh * NS * NDK + (size_t)qt * 16 * NDK;
    float*       Ab = attn + bh * (size_t)NS * NS + (size_t)qt * 16 * NS;

    const float scale = 0.125f;  // 1/sqrt(64)

    // ---- Q fragments for all 16 K-dim steps (A-matrix 16x4 layout) ----
    v2f qreg[16];
#pragma unroll
    for (int kk = 0; kk < 16; ++kk)
        qreg[kk] = *(const v2f*)(Qb + n * NDK + 4 * kk + 2 * lh);

    // per-lane running softmax stats for rows M = v + 8*lh
    float pm[8], ps[8];
#pragma unroll
    for (int v = 0; v < 8; ++v) { pm[v] = -1e30f; ps[v] = 0.0f; }

    // ================= Pass 1: row max / sum-of-exp =================
    for (int kt = 0; kt < NS / 16; ++kt) {
        v8f s = {0.f, 0.f, 0.f, 0.f, 0.f, 0.f, 0.f, 0.f};
        const float* Krow = Kb + (size_t)(kt * 16 + n) * NDK + 2 * lh;
#pragma unroll
        for (int kk = 0; kk < 16; ++kk) {
            v2f bf = *(const v2f*)(Krow + 4 * kk);
            s = __builtin_amdgcn_wmma_f32_16x16x4_f32(false, qreg[kk], false, bf,
                                                      (short)0, s, false, false);
        }
        const int kcol = kt * 16 + n;
#pragma unroll
        for (int v = 0; v < 8; ++v) {
            const int qrow = qt * 16 + v + 8 * lh;
            const int mok  = Mb[(size_t)qrow * NS + kcol];
            const float sv = mok ? s[v] * scale : -1e9f;
            const float nm = fmaxf(pm[v], sv);
            ps[v] = ps[v] * __expf(pm[v] - nm) + __expf(sv - nm);
            pm[v] = nm;
        }
    }

    // merge (m, l) across the 16 lanes of each half (log-sum-exp combine)
#pragma unroll
    for (int v = 0; v < 8; ++v) {
        float m = pm[v], l = ps[v];
#pragma unroll
        for (int off = 1; off < 16; off <<= 1) {
            const float om = __shfl_xor(m, off, 32);
            const float ol = __shfl_xor(l, off, 32);
            const float nm = fmaxf(m, om);
            l = l * __expf(m - nm) + ol * __expf(om - nm);
            m = nm;
        }
        pm[v] = m;
        ps[v] = 1.0f / l;          // keep reciprocal of the row sum
    }

    // ================= Pass 2: write attn, accumulate O = P @ V =====
    v8f acc[4];
#pragma unroll
    for (int nt = 0; nt < 4; ++nt)
        acc[nt] = (v8f){0.f, 0.f, 0.f, 0.f, 0.f, 0.f, 0.f, 0.f};

    for (int kt = 0; kt < NS / 16; ++kt) {
        // recompute score tile (K tiles are L0/L2 resident)
        v8f s = {0.f, 0.f, 0.f, 0.f, 0.f, 0.f, 0.f, 0.f};
        const float* Krow = Kb + (size_t)(kt * 16 + n) * NDK + 2 * lh;
#pragma unroll
        for (int kk = 0; kk < 16; ++kk) {
            v2f bf = *(const v2f*)(Krow + 4 * kk);
            s = __builtin_amdgcn_wmma_f32_16x16x4_f32(false, qreg[kk], false, bf,
                                                      (short)0, s, false, false);
        }
        const int kcol = kt * 16 + n;
        float p[8];
#pragma unroll
        for (int v = 0; v < 8; ++v) {
            const int qrow = qt * 16 + v + 8 * lh;
            const int mok  = Mb[(size_t)qrow * NS + kcol];
            const float sv = mok ? s[v] * scale : -1e9f;
            p[v] = __expf(sv - pm[v]) * ps[v];          // normalized prob
            Ab[(size_t)(v + 8 * lh) * NS + kcol] = p[v]; // write attn once
        }

        // transpose P tile (C/D layout -> A layout) through LDS
        __syncthreads();
#pragma unroll
        for (int v = 0; v < 8; ++v)
            pbuf[(v + 8 * lh) * 16 + n] = p[v];
        __syncthreads();

        v2f afr[4];
#pragma unroll
        for (int kk2 = 0; kk2 < 4; ++kk2)
            afr[kk2] = *(const v2f*)&pbuf[n * 16 + 4 * kk2 + 2 * lh];

        // O(16x64) += P(16x16) @ V(16x64), 4 output tiles x 4 K-steps
#pragma unroll
        for (int nt = 0; nt < 4; ++nt) {
#pragma unroll
            for (int kk2 = 0; kk2 < 4; ++kk2) {
                v2f vf;
                const size_t vrow = (size_t)(kt * 16 + 4 * kk2 + 2 * lh) * NDK + nt * 16 + n;
                vf.x = Vb[vrow];
                vf.y = Vb[vrow + NDK];
                acc[nt] = __builtin_amdgcn_wmma_f32_16x16x4_f32(false, afr[kk2], false, vf,
                                                                (short)0, acc[nt], false, false);
            }
        }
    }

    // ---- store O ----
#pragma unroll
    for (int nt = 0; nt < 4; ++nt)
#pragma unroll
        for (int v = 0; v < 8; ++v)
            Ob[(size_t)(v + 8 * lh) * NDK + nt * 16 + n] = acc[nt][v];
}

extern "C" void kernel_launch(void* const* d_in, const int* in_sizes, int n_in,
                              void* d_out, int out_size, void* d_ws, size_t ws_size,
                              hipStream_t stream) {
    const float* Q    = (const float*)d_in[0];
    const float* K    = (const float*)d_in[1];
    const float* V    = (const float*)d_in[2];
    const int*   mask = (const int*)d_in[3];

    float* out  = (float*)d_out;                               // [B,H,S,DK]
    float* attn = out + (size_t)NB * NH * NS * NDK;            // [B,H,S,S]

    const int nblocks = NB * NH * (NS / 16);                   // 8192 waves
    sdpa_fp32_wmma_kernel<<<nblocks, 32, 0, stream>>>(Q, K, V, mask, out, attn);
}